// PatchNorm_4475355922834
// MI455X (gfx1250) — compile-verified
//
#include <hip/hip_runtime.h>
#include <stdint.h>

// PatchNorm training forward for MI455X (gfx1250, wave32).
// counting-sort tokens into per-cell lists -> per-cell sums via TDM
// gather-mode DMA (tensor_load_to_lds, TENSORcnt) -> closed-form Welford ->
// streaming normalize with async global->LDS staging (ASYNCcnt).

namespace {
constexpr int kB = 16, kS = 4096, kD = 256;
constexpr int kH = 32, kW = 32;
constexpr int kCells = 3 * kH * kW;        // 3072
constexpr int kTok   = kB * kS;            // 65536 (token ids fit in 16 bits)
constexpr float kEps = 0.1f;
constexpr size_t kMeanOffBytes = (size_t)(3 * kCells + kTok) * 4;  // 1KB aligned
constexpr unsigned kInvRelBytes = (unsigned)kCells * kD * 4u;      // invws - meanws
}  // namespace

typedef unsigned __attribute__((ext_vector_type(4))) v4u;
typedef unsigned __attribute__((ext_vector_type(8))) v8u;

__device__ __forceinline__ unsigned rfl(unsigned x) {
  return (unsigned)__builtin_amdgcn_readfirstlane((int)x);
}

__global__ void k_zero_cnt(int* __restrict__ cnt) {
  int i = blockIdx.x * blockDim.x + threadIdx.x;
  if (i < kCells) cnt[i] = 0;
}

__global__ void k_count(const int* __restrict__ pc, const int* __restrict__ ph,
                        const int* __restrict__ pw,
                        const unsigned char* __restrict__ pad,  // jax bool = 1 byte
                        int* __restrict__ cnt) {
  int t = blockIdx.x * blockDim.x + threadIdx.x;
  if (t >= kTok || pad[t]) return;
  int cell = (pc[t] * kH + ph[t]) * kW + pw[t];
  atomicAdd(&cnt[cell], 1);
}

__global__ void k_scan(const int* __restrict__ cnt, int* __restrict__ offs,
                       int* __restrict__ cursor) {
  if (blockIdx.x == 0 && threadIdx.x == 0) {
    int run = 0;
    for (int c = 0; c < kCells; ++c) {
      offs[c] = run;
      cursor[c] = run;
      run += cnt[c];
    }
  }
}

__global__ void k_fill(const int* __restrict__ pc, const int* __restrict__ ph,
                       const int* __restrict__ pw,
                       const unsigned char* __restrict__ pad,
                       int* __restrict__ cursor, int* __restrict__ list) {
  int t = blockIdx.x * blockDim.x + threadIdx.x;
  if (t >= kTok || pad[t]) return;
  int cell = (pc[t] * kH + ph[t]) * kW + pw[t];
  int pos = atomicAdd(&cursor[cell], 1);
  list[pos] = t;
}

// One block (256 thr) per cell. Batches of up to 16 token rows are gathered
// into LDS by a single TDM gather-mode descriptor: 16-bit row indices in D#
// groups 2/3, tile = 256 x nbatch elements, row stride 256, data_size = 4B.
// Wave 0 issues the TDM op (per-wave, EXEC ignored) and waits on TENSORcnt.
__global__ void k_cell(const float* __restrict__ patches,
                       const float* __restrict__ n_in,
                       const float* __restrict__ mean_in,
                       const float* __restrict__ m2_in,
                       const int* __restrict__ cnt, const int* __restrict__ offs,
                       const int* __restrict__ list,
                       float* __restrict__ meanws, float* __restrict__ invws) {
  __shared__ float smem[16 * kD];  // 16 gathered token rows (16KB)
  int cell = blockIdx.x;
  int d = threadIdx.x;
  int cc = cnt[cell];   // block-uniform value
  int st = offs[cell];
  float s = 0.f, q = 0.f;

  for (int b = 0; b < cc; b += 16) {
    int nbatch = cc - b;
    if (nbatch > 16) nbatch = 16;
    if (threadIdx.x < 32) {  // wave 0 only: build D# and issue one TDM gather
      // D# group 0: count=1, gather_mode=1 (16-bit indices), lds addr,
      // 57-bit global addr of tensor base (= patches), type=2 ("image").
      uint64_t gaddr = (uint64_t)(uintptr_t)patches;
      unsigned lds_b = rfl((unsigned)(uintptr_t)(&smem[0]));  // low32 = LDS offset
      v4u g0 = {0x80000001u, lds_b, rfl((unsigned)gaddr),
                rfl(((unsigned)(gaddr >> 32) & 0x01FFFFFFu) | 0x80000000u)};
      // D# group 1: data_size=2 (4B); tensor_dim0=256, tensor_dim1=65536;
      // tile_dim0=256, tile_dim1=nbatch; tensor_dim0_stride=256.
      unsigned nb = rfl((unsigned)nbatch);
      v8u g1 = {0x00020000u, 256u << 16, 0u, (256u << 16) | 1u,
                nb, 256u, 0u, 0u};
      // D# groups 2/3: up to 16 packed 16-bit row indices (token ids).
      unsigned w[8];
#pragma unroll
      for (int k = 0; k < 8; ++k) {
        int j0 = 2 * k, j1 = 2 * k + 1;
        unsigned lo = (j0 < nbatch) ? (unsigned)list[st + b + j0] : 0u;
        unsigned hi = (j1 < nbatch) ? (unsigned)list[st + b + j1] : 0u;
        w[k] = rfl(lo | (hi << 16));
      }
      v4u g2 = {w[0], w[1], w[2], w[3]};
      v4u g3 = {w[4], w[5], w[6], w[7]};
      asm volatile("tensor_load_to_lds %0, %1, %2, %3"
                   :: "s"(g0), "s"(g1), "s"(g2), "s"(g3) : "memory");
      __builtin_amdgcn_s_wait_tensorcnt(0);
    }
    __syncthreads();  // gathered rows visible to all 8 waves
    for (int j = 0; j < nbatch; ++j) {
      float p = smem[j * kD + d];
      s += p;
      q = fmaf(p, p, q);
    }
    __syncthreads();  // LDS reuse for next batch
  }

  float cf = (float)cc;
  float n_new = n_in[cell] + cf;
  float n_g = fmaxf(n_new, 1.f);
  size_t o = (size_t)cell * kD + d;
  float mo = mean_in[o];
  float mn = mo + (s - cf * mo) / n_g;
  float m2n = m2_in[o] + q - s * (mo + mn) + cf * mo * mn;
  float var = (n_new < 2.f) ? 1.f : (m2n / n_g);
  meanws[o] = mn;
  invws[o] = 1.f / (sqrtf(var) + kEps);
}

// One block per token: stage mean row + inv row (2KB) into LDS via per-lane
// CDNA5 async DMA (GVS mode: uniform SGPR base + 32-bit lane offset), wait on
// ASYNCcnt, then normalize.
__global__ void k_out(const float* __restrict__ patches,
                      const int* __restrict__ pc, const int* __restrict__ ph,
                      const int* __restrict__ pw,
                      const unsigned char* __restrict__ pad,
                      const float* __restrict__ meanws,  // invws = meanws + kInvRelBytes
                      float* __restrict__ out) {
  __shared__ float smem[2 * kD];  // [0,256): mean row, [256,512): inv row
  int t = blockIdx.x;
  int d = threadIdx.x;
  size_t o = (size_t)t * kD + d;
  if (pad[t]) {  // block-uniform; padded tokens output zeros
    out[o] = 0.f;
    return;
  }
  int cell = (pc[t] * kH + ph[t]) * kW + pw[t];
  if (d < 128) {  // waves 0..3, EXEC all-ones in each
    unsigned idx = (unsigned)(d & 63);
    unsigned gofs = (unsigned)cell * (unsigned)(kD * 4) +
                    ((d < 64) ? 0u : kInvRelBytes) + idx * 16u;
    unsigned lofs = (unsigned)(uintptr_t)(&smem[0]) + (unsigned)d * 16u;
    asm volatile("global_load_async_to_lds_b128 %0, %1, %2"
                 :: "v"(lofs), "v"(gofs), "s"(meanws) : "memory");
    asm volatile("s_wait_asynccnt 0" ::: "memory");
  }
  __syncthreads();
  float p = patches[o];
  out[o] = (p - smem[d]) * smem[kD + d];
}

extern "C" void kernel_launch(void* const* d_in, const int* in_sizes, int n_in_cnt,
                              void* d_out, int out_size, void* d_ws, size_t ws_size,
                              hipStream_t stream) {
  (void)in_sizes; (void)n_in_cnt; (void)out_size; (void)ws_size;
  const float* patches = (const float*)d_in[0];
  const float* n_in    = (const float*)d_in[1];
  const float* mean_in = (const float*)d_in[2];
  const float* m2_in   = (const float*)d_in[3];
  const int* pc = (const int*)d_in[4];
  const int* ph = (const int*)d_in[5];
  const int* pw = (const int*)d_in[6];
  const unsigned char* pad = (const unsigned char*)d_in[7];  // bool array
  float* out = (float*)d_out;

  char* ws = (char*)d_ws;
  int* cnt    = (int*)ws;
  int* offs   = cnt + kCells;
  int* cursor = offs + kCells;
  int* list   = cursor + kCells;
  float* meanws = (float*)(ws + kMeanOffBytes);
  float* invws  = meanws + (size_t)kCells * kD;

  k_zero_cnt<<<(kCells + 255) / 256, 256, 0, stream>>>(cnt);
  k_count<<<kTok / 256, 256, 0, stream>>>(pc, ph, pw, pad, cnt);
  k_scan<<<1, 32, 0, stream>>>(cnt, offs, cursor);
  k_fill<<<kTok / 256, 256, 0, stream>>>(pc, ph, pw, pad, cursor, list);
  k_cell<<<kCells, kD, 0, stream>>>(patches, n_in, mean_in, m2_in,
                                    cnt, offs, list, meanws, invws);
  k_out<<<kTok, kD, 0, stream>>>(patches, pc, ph, pw, pad, meanws, out);
}